// MPM_6622839571145
// MI455X (gfx1250) — compile-verified
//
#include <hip/hip_runtime.h>
#include <hip/hip_bf16.h>

// ---------------------------------------------------------------------------
// MPM crowd sim on gfx1250. Dense W[G,P] contractions are reformulated as
// fp32 GEMMs executed with V_WMMA_F32_16X16X4_F32, with the B-spline weight
// matrix generated on the fly in registers (A operand) and split-K partial
// sums reduced via global_atomic_add_f32. Two independent WMMA accumulator
// chains per wave hide the D->C dependency. The O(P^2) pairwise force is
// split 8-ways over j with atomic reduction into the feature matrix.
// All state lives in d_ws (~1.4 MB -> resident in the 192 MB L2).
// ---------------------------------------------------------------------------

typedef __attribute__((ext_vector_type(2))) float v2f;
typedef __attribute__((ext_vector_type(8))) float v8f;

namespace mpm {

constexpr int   N_P     = 4000;
constexpr int   NG      = 64;
constexpr int   N_GRIDS = NG * NG;     // 4096
constexpr int   N_SUB   = 4;
constexpr float DX      = 1.0f / 64.0f;
constexpr float INV_DX  = 64.0f;
constexpr float DT      = 2e-3f;
constexpr float P_MASS  = 1.0f;
constexpr float P_VOL   = (DX * 0.5f) * (DX * 0.5f);
constexpr float E_SCALE = 100.0f;
constexpr float K_SCALE = 5.0f;
constexpr float THETA1  = 0.02f;
constexpr float THETA2  = 0.06f;
constexpr float COMFORT = 0.05f;
constexpr float W_EXT   = 1.0f;
constexpr float GOALX   = 0.5f, GOALY = 0.5f;
constexpr float BOUND   = 3.0f * DX;

constexpr int NF = 16;                 // padded feature columns

constexpr int KSPLIT_P2G = 10;         // 10 * 400 = 4000 particles
constexpr int KCHUNK_P2G = N_P / KSPLIT_P2G;
constexpr int KSPLIT_G2P = 16;         // 16 * 256 = 4096 cells
constexpr int KCHUNK_G2P = N_GRIDS / KSPLIT_G2P;
constexpr int JSPLIT     = 8;          // pairwise j-range splits
constexpr int JCHUNK     = N_P / JSPLIT;   // 500

static_assert(KCHUNK_P2G % 8 == 0, "P2G K chunk must divide 8-wide unroll");
static_assert(KCHUNK_G2P % 8 == 0, "G2P K chunk must divide 8-wide unroll");
static_assert(N_P % 16 == 0 && N_GRIDS % 16 == 0, "tile sizes");
static_assert(N_P % JSPLIT == 0, "j split");

__device__ __forceinline__ float bspline(float d) {
  float a = fabsf(d) * INV_DX;
  float t = 1.5f - a;
  float r = 0.5f * t * t;
  r = (a < 0.5f) ? (0.75f - a * a) : r;
  r = (a < 1.5f) ? r : 0.0f;
  return r;
}

__device__ __forceinline__ float sigmoidf(float z) {
  return 1.0f / (1.0f + __expf(-z));
}

// ---------------------------------------------------------------------------
// Copy inputs to evolving state buffers; compute fixed per-particle d_vel.
// ---------------------------------------------------------------------------
__global__ void init_state(const float* __restrict__ pos_in,
                           const float* __restrict__ vel_in,
                           const float* __restrict__ C_in,
                           const float* __restrict__ J_in,
                           float* __restrict__ pos, float* __restrict__ vel,
                           float* __restrict__ C, float* __restrict__ J,
                           float* __restrict__ d_vel) {
  int p = blockIdx.x * blockDim.x + threadIdx.x;
  if (p >= N_P) return;
  float vx = vel_in[2 * p], vy = vel_in[2 * p + 1];
  pos[2 * p]     = pos_in[2 * p];
  pos[2 * p + 1] = pos_in[2 * p + 1];
  vel[2 * p]     = vx;
  vel[2 * p + 1] = vy;
#pragma unroll
  for (int k = 0; k < 4; ++k) C[4 * p + k] = C_in[4 * p + k];
  J[p] = J_in[p];
  float dv2 = vx * vx + vy * vy;
  d_vel[p] = sqrtf(dv2 > 0.0f ? dv2 : 1e-12f);
}

// ---------------------------------------------------------------------------
// Per-particle heads (E, K), goal force, stress, and the 16-column P2G
// feature matrix X[P,16]. Pairwise force columns (11/12) are seeded with the
// external force; the pair_forces kernel atomically accumulates on top.
// ---------------------------------------------------------------------------
__global__ __launch_bounds__(256) void particle_features(
    const float* __restrict__ pos, const float* __restrict__ vel,
    const float* __restrict__ C, const float* __restrict__ J,
    const float* __restrict__ d_vel,
    const float* __restrict__ We, const float* __restrict__ be,
    const float* __restrict__ Wk, const float* __restrict__ bk,
    float* __restrict__ X, float* __restrict__ Kbuf) {
  int p = blockIdx.x * 256 + threadIdx.x;
  if (p >= N_P) return;
  float px = pos[2 * p], py = pos[2 * p + 1];
  float vx = vel[2 * p], vy = vel[2 * p + 1];

  // sigmoid-linear heads on [pos, vel]
  float zE = px * We[0] + py * We[1] + vx * We[2] + vy * We[3] + be[0];
  float zK = px * Wk[0] + py * Wk[1] + vx * Wk[2] + vy * Wk[3] + bk[0];
  float E  = E_SCALE * sigmoidf(zE);
  float Kp = K_SCALE * sigmoidf(zK);
  Kbuf[p] = Kp;

  // external goal-seeking force (DT cancels algebraically)
  float gx = GOALX - px, gy = GOALY - py;
  float g2 = gx * gx + gy * gy;
  float gdis = sqrtf(g2 > 0.0f ? g2 : 1.0f);
  float dvl = d_vel[p];
  float extx = (dvl * gx / gdis - vx) * P_MASS * W_EXT;
  float exty = (dvl * gy / gdis - vy) * P_MASS * W_EXT;

  float Jp = J[p];
  float stress = -DT * 4.0f * INV_DX * INV_DX * E * P_VOL * (Jp - 1.0f);
  float c00 = C[4 * p + 0], c01 = C[4 * p + 1];
  float c10 = C[4 * p + 2], c11 = C[4 * p + 3];

  float* x = X + p * NF;
  x[0]  = P_MASS * vx;                 // momentum
  x[1]  = P_MASS * vy;
  x[2]  = P_MASS * c00;                // affine C (grid-side gp term)
  x[3]  = P_MASS * c01;
  x[4]  = P_MASS * c10;
  x[5]  = P_MASS * c11;
  x[6]  = P_MASS * (c00 * px + c01 * py);   // C . pos term
  x[7]  = P_MASS * (c10 * px + c11 * py);
  x[8]  = stress;                      // elastic (grid-side gp term)
  x[9]  = stress * px;
  x[10] = stress * py;
  x[11] = extx;                        // + pairwise force (atomic, next kernel)
  x[12] = exty;
  x[13] = 1.0f;                        // mass accumulation
  x[14] = 0.0f;
  x[15] = 0.0f;
}

// ---------------------------------------------------------------------------
// O(P^2) repulsion, split JSPLIT ways over the j-range for parallelism.
// Partial (fx,fy) reduced atomically into X columns 11/12.
// ---------------------------------------------------------------------------
__global__ __launch_bounds__(256) void pair_forces(
    const float* __restrict__ pos, const float* __restrict__ Kbuf,
    float* __restrict__ X) {
  __shared__ float spx[256];
  __shared__ float spy[256];
  int p = blockIdx.x * 256 + threadIdx.x;
  bool valid = p < N_P;
  float px = 0.f, py = 0.f, Kp = 0.f;
  if (valid) {
    px = pos[2 * p]; py = pos[2 * p + 1];
    Kp = Kbuf[p];
  }
  float fx = 0.f, fy = 0.f;
  const int jbeg = blockIdx.y * JCHUNK;
  const int jend = jbeg + JCHUNK;
  for (int base = jbeg; base < jend; base += 256) {
    int j = base + threadIdx.x;
    float qx = 1e9f, qy = 1e9f;  // "far away": never interacts
    if (j < jend) { qx = pos[2 * j]; qy = pos[2 * j + 1]; }
    __syncthreads();
    spx[threadIdx.x] = qx;
    spy[threadIdx.x] = qy;
    __syncthreads();
    int cnt = min(256, jend - base);
    for (int t = 0; t < cnt; ++t) {
      float dx = px - spx[t];
      float dy = py - spy[t];
      float d2 = dx * dx + dy * dy;
      float dis = sqrtf(d2 > 0.0f ? d2 : 1.0f);
      bool  msk = (d2 > 0.0f) && (dis < THETA2);
      float nd  = fmaxf(dis, THETA1 + 1e-5f);
      float tmp = Kp * (-__logf((nd - THETA1) * (1.0f / COMFORT)) * DT);
      float s   = msk ? (tmp / dis) : 0.0f;
      fx += s * dx;
      fy += s * dy;
    }
  }
  if (valid) {
    atomicAdd(&X[p * NF + 11], fx);
    atomicAdd(&X[p * NF + 12], fy);
  }
}

// ---------------------------------------------------------------------------
// P2G as fp32 WMMA GEMM:  D[G,16] += W[G,P] @ X[P,16]
// A (16 grid cells x 4 particles) generated in registers from positions.
// f32 16x16x4 A layout: lanes 0-15 hold K=0/1 in v[0]/v[1]; lanes 16-31 K=2/3.
// Two independent accumulator chains (8-wide K unroll) hide WMMA latency.
// ---------------------------------------------------------------------------
__global__ __launch_bounds__(32) void p2g_wmma(const float* __restrict__ X,
                                               const float* __restrict__ pos,
                                               float* __restrict__ D) {
  const int tile = blockIdx.x;       // grid-cell tile (16 rows)
  const int ks   = blockIdx.y;       // split-K slice
  const int lane = threadIdx.x;      // wave32
  const int m    = lane & 15;
  const int hi   = lane >> 4;        // 0: K pair {0,1}; 1: K pair {2,3}
  const int n    = lane & 15;

  const int g = tile * 16 + m;
  const float gpx = (float)(g >> 6) * DX;
  const float gpy = (float)(g & 63) * DX;

  const float2* pos2 = (const float2*)pos;
  v8f c0 = {};  // two independent fp32 accumulator tiles
  v8f c1 = {};

  const int kbeg = ks * KCHUNK_P2G;
  const int kend = kbeg + KCHUNK_P2G;
  for (int k = kbeg; k < kend; k += 8) {
    __builtin_prefetch(&X[(k + 16) * NF + n], 0, 0);  // global_prefetch_b8
    const int ka = k + hi * 2;
    float2 pa0 = pos2[ka];
    float2 pa1 = pos2[ka + 1];
    const int kb = k + 4 + hi * 2;
    float2 pb0 = pos2[kb];
    float2 pb1 = pos2[kb + 1];
    v2f a0, b0, a1, b1;
    a0.x = bspline(gpx - pa0.x) * bspline(gpy - pa0.y);
    a0.y = bspline(gpx - pa1.x) * bspline(gpy - pa1.y);
    b0.x = X[ka * NF + n];
    b0.y = X[(ka + 1) * NF + n];
    a1.x = bspline(gpx - pb0.x) * bspline(gpy - pb0.y);
    a1.y = bspline(gpx - pb1.x) * bspline(gpy - pb1.y);
    b1.x = X[kb * NF + n];
    b1.y = X[(kb + 1) * NF + n];
    c0 = __builtin_amdgcn_wmma_f32_16x16x4_f32(false, a0, false, b0,
                                               (short)0, c0, false, false);
    c1 = __builtin_amdgcn_wmma_f32_16x16x4_f32(false, a1, false, b1,
                                               (short)0, c1, false, false);
  }
  // C/D layout: VGPR r -> rows (r, r+8); lane halves select row group
#pragma unroll
  for (int r = 0; r < 8; ++r) {
    int gg = tile * 16 + r + hi * 8;
    atomicAdd(&D[gg * NF + n], c0[r] + c1[r]);
  }
}

// ---------------------------------------------------------------------------
// Grid epilogue: combine GEMM columns into grid momentum, apply sigmoid gate,
// emit grid_v_out slice of d_out and the 6-column G2P feature matrix Y[G,16].
// ---------------------------------------------------------------------------
__global__ void grid_epilogue(const float* __restrict__ D,
                              const float* __restrict__ Wa,
                              const float* __restrict__ ba,
                              float* __restrict__ out_gvo,
                              float* __restrict__ Y) {
  int g = blockIdx.x * blockDim.x + threadIdx.x;
  if (g >= N_GRIDS) return;
  const float* d = D + g * NF;
  float gpx = (float)(g >> 6) * DX;
  float gpy = (float)(g & 63) * DX;

  // w_mom + w_ang
  float momx = d[0] + (gpx * d[2] + gpy * d[3] - d[6]);
  float momy = d[1] + (gpx * d[4] + gpy * d[5] - d[7]);
  // w_el
  float elx = gpx * d[8] - d[9];
  float ely = gpy * d[8] - d[10];

  float mass = d[13] * P_MASS;
  bool  gm   = mass > 0.0f;
  float invm = 1.0f / (gm ? mass : 1.0f);
  float vinx = gm ? momx * invm : 0.0f;
  float viny = gm ? momy * invm : 0.0f;
  float alpha = gm ? sigmoidf(vinx * Wa[0] + viny * Wa[1] + ba[0]) : 0.0f;

  float gmx = momx + elx + d[11] + momx * alpha * DT;
  float gmy = momy + ely + d[12] + momy * alpha * DT;
  float gvx = gm ? gmx * invm : 0.0f;
  float gvy = gm ? gmy * invm : 0.0f;

  out_gvo[2 * g]     = gvx;
  out_gvo[2 * g + 1] = gvy;

  float* y = Y + g * NF;
  y[0] = gvx;        // vel_n
  y[1] = gvy;
  y[2] = gvx * gpx;  // S[d][e] = sum_g w * gvo_d * gp_e
  y[3] = gvx * gpy;
  y[4] = gvy * gpx;
  y[5] = gvy * gpy;
#pragma unroll
  for (int t = 6; t < NF; ++t) y[t] = 0.0f;
}

// ---------------------------------------------------------------------------
// G2P as fp32 WMMA GEMM:  D2[P,16] += W^T[P,G] @ Y[G,16]
// A (16 particles x 4 grid cells) generated in registers; dual accumulators.
// ---------------------------------------------------------------------------
__global__ __launch_bounds__(32) void g2p_wmma(const float* __restrict__ Y,
                                               const float* __restrict__ pos,
                                               float* __restrict__ D2) {
  const int tile = blockIdx.x;       // particle tile (16 rows)
  const int ks   = blockIdx.y;       // split-K slice over grid cells
  const int lane = threadIdx.x;
  const int m    = lane & 15;
  const int hi   = lane >> 4;
  const int n    = lane & 15;

  const int p = tile * 16 + m;       // always < N_P (250*16 == 4000)
  const float2 pp = ((const float2*)pos)[p];

  v8f c0 = {};
  v8f c1 = {};
  const int kbeg = ks * KCHUNK_G2P;
  const int kend = kbeg + KCHUNK_G2P;
  for (int k = kbeg; k < kend; k += 8) {
    __builtin_prefetch(&Y[(k + 16) * NF + n], 0, 0);  // global_prefetch_b8
    const int ka = k + hi * 2;
    const int kb = k + 4 + hi * 2;
    float ga0x = (float)(ka >> 6) * DX, ga0y = (float)(ka & 63) * DX;
    float ga1x = (float)((ka + 1) >> 6) * DX, ga1y = (float)((ka + 1) & 63) * DX;
    float gb0x = (float)(kb >> 6) * DX, gb0y = (float)(kb & 63) * DX;
    float gb1x = (float)((kb + 1) >> 6) * DX, gb1y = (float)((kb + 1) & 63) * DX;
    v2f a0, b0, a1, b1;
    a0.x = bspline(ga0x - pp.x) * bspline(ga0y - pp.y);
    a0.y = bspline(ga1x - pp.x) * bspline(ga1y - pp.y);
    b0.x = Y[ka * NF + n];
    b0.y = Y[(ka + 1) * NF + n];
    a1.x = bspline(gb0x - pp.x) * bspline(gb0y - pp.y);
    a1.y = bspline(gb1x - pp.x) * bspline(gb1y - pp.y);
    b1.x = Y[kb * NF + n];
    b1.y = Y[(kb + 1) * NF + n];
    c0 = __builtin_amdgcn_wmma_f32_16x16x4_f32(false, a0, false, b0,
                                               (short)0, c0, false, false);
    c1 = __builtin_amdgcn_wmma_f32_16x16x4_f32(false, a1, false, b1,
                                               (short)0, c1, false, false);
  }
#pragma unroll
  for (int r = 0; r < 8; ++r) {
    int ppr = tile * 16 + r + hi * 8;
    atomicAdd(&D2[ppr * NF + n], c0[r] + c1[r]);
  }
}

// ---------------------------------------------------------------------------
// Particle update: vel_n, C_n, J_n, clamped pos_n -> next-substep buffers.
// ---------------------------------------------------------------------------
__global__ void particle_update(const float* __restrict__ D2,
                                const float* __restrict__ pos,
                                const float* __restrict__ J,
                                float* __restrict__ pos_n,
                                float* __restrict__ vel_n,
                                float* __restrict__ C_n,
                                float* __restrict__ J_n) {
  int p = blockIdx.x * blockDim.x + threadIdx.x;
  if (p >= N_P) return;
  const float* d = D2 + p * NF;
  float vnx = d[0], vny = d[1];
  float px = pos[2 * p], py = pos[2 * p + 1];
  const float k4 = 4.0f * INV_DX * INV_DX;
  float c00 = k4 * (d[2] - px * vnx);
  float c01 = k4 * (d[3] - py * vnx);
  float c10 = k4 * (d[4] - px * vny);
  float c11 = k4 * (d[5] - py * vny);
  float Jn = J[p] * (1.0f + DT * (c00 + c11));
  float pnx = fminf(fmaxf(px + DT * vnx, BOUND), 1.0f - BOUND);
  float pny = fminf(fmaxf(py + DT * vny, BOUND), 1.0f - BOUND);
  pos_n[2 * p]     = pnx;
  pos_n[2 * p + 1] = pny;
  vel_n[2 * p]     = vnx;
  vel_n[2 * p + 1] = vny;
  C_n[4 * p + 0] = c00;
  C_n[4 * p + 1] = c01;
  C_n[4 * p + 2] = c10;
  C_n[4 * p + 3] = c11;
  J_n[p] = Jn;
}

}  // namespace mpm

// ---------------------------------------------------------------------------
// Host orchestration (graph-capture safe: launches + hipMemsetAsync only).
// ---------------------------------------------------------------------------
extern "C" void kernel_launch(void* const* d_in, const int* in_sizes, int n_in,
                              void* d_out, int out_size, void* d_ws,
                              size_t ws_size, hipStream_t stream) {
  using namespace mpm;
  const float* pos_in = (const float*)d_in[0];
  const float* vel_in = (const float*)d_in[1];
  const float* C_in   = (const float*)d_in[2];
  const float* J_in   = (const float*)d_in[3];
  const float* We     = (const float*)d_in[4];
  const float* be     = (const float*)d_in[5];
  const float* Wk     = (const float*)d_in[6];
  const float* bk     = (const float*)d_in[7];
  const float* Wa     = (const float*)d_in[8];
  const float* ba     = (const float*)d_in[9];
  float* out = (float*)d_out;

  // carve workspace (float units)
  float* w = (float*)d_ws;
  float* posb[2] = {w, w + 2 * N_P};            w += 4 * N_P;
  float* velb[2] = {w, w + 2 * N_P};            w += 4 * N_P;
  float* Cb[2]   = {w, w + 4 * N_P};            w += 8 * N_P;
  float* Jb[2]   = {w, w + N_P};                w += 2 * N_P;
  float* dvel    = w;                           w += N_P;
  float* Kbuf    = w;                           w += N_P;
  float* X       = w;                           w += N_P * NF;
  float* D       = w;                           w += N_GRIDS * NF;
  float* Y       = w;                           w += N_GRIDS * NF;
  float* D2      = w;                           w += N_P * NF;

  const int PB = (N_P + 255) / 256;   // 16 blocks
  const int GB = (N_GRIDS + 255) / 256;

  init_state<<<PB, 256, 0, stream>>>(pos_in, vel_in, C_in, J_in,
                                     posb[0], velb[0], Cb[0], Jb[0], dvel);

  int cur = 0;
  for (int s = 0; s < N_SUB; ++s) {
    int nxt = cur ^ 1;
    particle_features<<<PB, 256, 0, stream>>>(posb[cur], velb[cur], Cb[cur],
                                              Jb[cur], dvel, We, be, Wk, bk,
                                              X, Kbuf);
    pair_forces<<<dim3(PB, JSPLIT), 256, 0, stream>>>(posb[cur], Kbuf, X);
    hipMemsetAsync(D, 0, (size_t)N_GRIDS * NF * sizeof(float), stream);
    p2g_wmma<<<dim3(N_GRIDS / 16, KSPLIT_P2G), 32, 0, stream>>>(X, posb[cur], D);
    grid_epilogue<<<GB, 256, 0, stream>>>(D, Wa, ba,
                                          out + (size_t)s * N_GRIDS * 2, Y);
    hipMemsetAsync(D2, 0, (size_t)N_P * NF * sizeof(float), stream);
    g2p_wmma<<<dim3(N_P / 16, KSPLIT_G2P), 32, 0, stream>>>(Y, posb[cur], D2);
    particle_update<<<PB, 256, 0, stream>>>(D2, posb[cur], Jb[cur],
                                            posb[nxt], velb[nxt], Cb[nxt],
                                            Jb[nxt]);
    cur = nxt;
  }
}